// EmbeddingKWGCN_21878563406447
// MI455X (gfx1250) — compile-verified
//
#include <hip/hip_runtime.h>

typedef __attribute__((ext_vector_type(2))) float v2f;
typedef __attribute__((ext_vector_type(8))) float v8f;

#define FIN  32
#define HDIM 32

// ---------------------------------------------------------------------------
// Zero a float buffer (workspace is poisoned by the harness each run).
// ---------------------------------------------------------------------------
__global__ void zero_f32(float* __restrict__ p, int n) {
  int i = blockIdx.x * blockDim.x + threadIdx.x;
  int stride = gridDim.x * blockDim.x;
  for (; i < n; i += stride) p[i] = 0.0f;
}

// ---------------------------------------------------------------------------
// SpMM scatter: out[(t*N+row)*32 + f] += val * X[(t*N+col)*32 + f]
// One thread per (nnz, 4-feature chunk): float4 gather + 4 f32 atomics.
// Atomics land in L2 (AX is ~20.5MB, L2-resident on the 192MB L2).
// ---------------------------------------------------------------------------
__global__ void __launch_bounds__(256)
spmm_scatter(const float* __restrict__ Xf,
             const int*   __restrict__ idx,   // stacked (3, nnz): t, row, col
             const float* __restrict__ vals,
             float*       __restrict__ outAX,
             int nnz, int N) {
  int tid = blockIdx.x * blockDim.x + threadIdx.x;
  int e = tid >> 3;
  if (e >= nnz) return;
  int fq = (tid & 7) * 4;
  int t = idx[e];
  int r = idx[nnz + e];
  int c = idx[2 * nnz + e];
  float v = vals[e];
  const float4 x = *(const float4*)(Xf + (size_t)(t * N + c) * FIN + fq);
  float* dst = outAX + (size_t)(t * N + r) * FIN + fq;
  atomicAdd(dst + 0, v * x.x);
  atomicAdd(dst + 1, v * x.y);
  atomicAdd(dst + 2, v * x.z);
  atomicAdd(dst + 3, v * x.w);
}

// ---------------------------------------------------------------------------
// Dense GEMM: O[rows,32] = (relu?)( A[rows,32] @ W[32,32] )
// One wave per 16-row tile; two 16x16 N-tiles; K=32 as 8x V_WMMA_F32_16X16X4_F32.
// Fragment layouts per CDNA5 ISA 7.12.2:
//   A 16x4 f32 : lanes 0-15 hold K={0,1}, lanes 16-31 hold K={2,3} of row lane&15
//   B 4x16 f32 : lanes 0-15 hold K-rows {0,1}, lanes 16-31 hold {2,3}, col lane&15
//   C/D 16x16  : VGPR r -> row r + (lane<16?0:8), col lane&15
// ---------------------------------------------------------------------------
template <bool RELU>
__global__ void __launch_bounds__(256)
gemm32_wmma(const float* __restrict__ A,
            const float* __restrict__ W,
            float*       __restrict__ O,
            int rows) {
  const int lane = threadIdx.x & 31;
  const int wave = threadIdx.x >> 5;
  const int tile = blockIdx.x * (blockDim.x >> 5) + wave;
  const int tiles = rows >> 4;
  if (tile >= tiles) return;

  const int m    = lane & 15;
  const int klo  = (lane < 16) ? 0 : 2;
  const int col  = lane & 15;
  const float* arow = A + (size_t)(tile * 16 + m) * FIN;

  v8f acc0 = {};
  v8f acc1 = {};
#pragma unroll
  for (int kk = 0; kk < FIN; kk += 4) {
    const int k = kk + klo;
    v2f a;  a.x  = arow[k];                  a.y  = arow[k + 1];
    v2f b0; b0.x = W[k * HDIM + col];        b0.y = W[(k + 1) * HDIM + col];
    v2f b1; b1.x = W[k * HDIM + 16 + col];   b1.y = W[(k + 1) * HDIM + 16 + col];
    acc0 = __builtin_amdgcn_wmma_f32_16x16x4_f32(false, a, false, b0, (short)0, acc0, false, false);
    acc1 = __builtin_amdgcn_wmma_f32_16x16x4_f32(false, a, false, b1, (short)0, acc1, false, false);
  }

  const int half = (lane < 16) ? 0 : 8;
#pragma unroll
  for (int r = 0; r < 8; ++r) {
    float v0 = acc0[r], v1 = acc1[r];
    if (RELU) { v0 = fmaxf(v0, 0.0f); v1 = fmaxf(v1, 0.0f); }
    const size_t orow = (size_t)(tile * 16 + r + half);
    O[orow * HDIM + col]      = v0;
    O[orow * HDIM + 16 + col] = v1;
  }
}

// ---------------------------------------------------------------------------
// Edge head: out[e] = Z[src(e)] @ U[:32] + Z[trg(e)] @ U[32:]
// Batched as WMMA: 16 edges per wave, K=64 concat feature vs U[64,32].
// Gathers hit L2 (Z is 20.5MB); output write (128MB) is the HBM floor.
// ---------------------------------------------------------------------------
__global__ void __launch_bounds__(256)
edge_head_wmma(const float* __restrict__ Z,
               const int*   __restrict__ edges,  // stacked (3, ne): t, src, trg
               const float* __restrict__ U,      // [64, 32]
               float*       __restrict__ out,    // [ne, 32]
               int ne, int N) {
  const int lane = threadIdx.x & 31;
  const int wave = threadIdx.x >> 5;
  const int tile = blockIdx.x * (blockDim.x >> 5) + wave;
  const int tiles = (ne + 15) >> 4;
  if (tile >= tiles) return;

  const int m = lane & 15;
  int e = tile * 16 + m;
  if (e >= ne) e = ne - 1;  // clamp reads; stores masked below (EXEC stays full)
  const int t = edges[e];
  const int s = edges[ne + e];
  const int d = edges[2 * ne + e];
  const float* zs = Z + (size_t)(t * N + s) * HDIM;
  const float* zt = Z + (size_t)(t * N + d) * HDIM;

  const int klo = (lane < 16) ? 0 : 2;
  const int col = lane & 15;

  v8f acc0 = {};
  v8f acc1 = {};
#pragma unroll
  for (int kk = 0; kk < 2 * HDIM; kk += 4) {
    const int k = kk + klo;                       // k and k+1 stay on same side of 32
    const float* p = (k < HDIM) ? (zs + k) : (zt + (k - HDIM));
    v2f a;  a.x  = p[0];                     a.y  = p[1];
    v2f b0; b0.x = U[k * HDIM + col];        b0.y = U[(k + 1) * HDIM + col];
    v2f b1; b1.x = U[k * HDIM + 16 + col];   b1.y = U[(k + 1) * HDIM + 16 + col];
    acc0 = __builtin_amdgcn_wmma_f32_16x16x4_f32(false, a, false, b0, (short)0, acc0, false, false);
    acc1 = __builtin_amdgcn_wmma_f32_16x16x4_f32(false, a, false, b1, (short)0, acc1, false, false);
  }

  const int half = (lane < 16) ? 0 : 8;
#pragma unroll
  for (int r = 0; r < 8; ++r) {
    const int eo = tile * 16 + r + half;
    if (eo < ne) {
      out[(size_t)eo * HDIM + col]      = acc0[r];
      out[(size_t)eo * HDIM + 16 + col] = acc1[r];
    }
  }
}

// ---------------------------------------------------------------------------
// Host-side orchestration (graph-capture safe: only kernel launches on stream)
// ---------------------------------------------------------------------------
extern "C" void kernel_launch(void* const* d_in, const int* in_sizes, int n_in,
                              void* d_out, int out_size, void* d_ws, size_t ws_size,
                              hipStream_t stream) {
  const float* X     = (const float*)d_in[0];   // (T, N, 32)
  const float* Avals = (const float*)d_in[1];   // (NNZ,)
  const float* W1    = (const float*)d_in[2];   // (32, 32)
  const float* W2    = (const float*)d_in[3];   // (32, 32)
  const float* U     = (const float*)d_in[4];   // (64, 32)
  const int*   Aidx  = (const int*)d_in[5];     // (3, NNZ)
  const int*   edges = (const int*)d_in[6];     // (3, NE)

  const int TN  = in_sizes[0] / FIN;  // T * N rows
  const int nnz = in_sizes[1];
  const int ne  = in_sizes[6] / 3;
  const int N   = TN / 8;             // T = 8 in the reference

  float* B0  = (float*)d_ws;                    // spmm outputs (AX, then A@Y)
  float* B1  = B0 + (size_t)TN * HDIM;          // Y, then Z
  float* out = (float*)d_out;

  const int nelt      = TN * HDIM;
  const int zblocks   = (nelt + 255) / 256;
  const int sblocks   = (nnz * 8 + 255) / 256;
  const int gtiles    = TN / 16;
  const int gblocks   = (gtiles + 7) / 8;       // 8 waves / block
  const int etiles    = (ne + 15) / 16;
  const int eblocks   = (etiles + 7) / 8;

  // Layer 1: AX = A @ X ; Y = relu(AX @ W1)
  zero_f32<<<zblocks, 256, 0, stream>>>(B0, nelt);
  spmm_scatter<<<sblocks, 256, 0, stream>>>(X, Aidx, Avals, B0, nnz, N);
  gemm32_wmma<true><<<gblocks, 256, 0, stream>>>(B0, W1, B1, TN);

  // Layer 2: S = A @ Y ; Z = S @ W2
  zero_f32<<<zblocks, 256, 0, stream>>>(B0, nelt);
  spmm_scatter<<<sblocks, 256, 0, stream>>>(B1, Aidx, Avals, B0, nnz, N);
  gemm32_wmma<false><<<gblocks, 256, 0, stream>>>(B0, W2, B1, TN);

  // Edge head: out = Z[src] @ U[:32] + Z[trg] @ U[32:]
  edge_head_wmma<<<eblocks, 256, 0, stream>>>(B1, edges, U, out, ne, N);
}